// FrustumToVoxel_4440996184155
// MI455X (gfx1250) — compile-verified
//
#include <hip/hip_runtime.h>

typedef __attribute__((ext_vector_type(2))) float v2f;
typedef __attribute__((ext_vector_type(8))) float v8f;

namespace {
constexpr int B = 2, T = 3, NCAM = 6, C = 16;
constexpr int FD = 48, FH = 24, FW = 64;
constexpr int GX = 100, GY = 100, GZ = 8;
constexpr int OUT_CH = 40;
constexpr int NPT = GX * GY * GZ;       // 80000 voxels
constexpr int K1 = NCAM * C;            // 96
constexpr int K2 = T * OUT_CH;          // 120
constexpr int FSTRIDE = FD * FH * FW;   // 73728
constexpr int PTILES = NPT / 16;        // 5000
}

// ---------------------------------------------------------------------------
// Kernel 0: cumulative warp matrices (reverse-time product of DoF 4x4s).
// cum[b][t] stores first 3 rows (12 floats) of the 4x4 cumulative transform.
// ---------------------------------------------------------------------------
__global__ void k_cum(const float* __restrict__ DoF, float* __restrict__ cum) {
  int b = threadIdx.x;
  if (b >= B) return;
  float c[16] = {1.f,0.f,0.f,0.f, 0.f,1.f,0.f,0.f,
                 0.f,0.f,1.f,0.f, 0.f,0.f,0.f,1.f};
  for (int t = T - 1; t >= 0; --t) {
    if (t < T - 1) {
      const float* d = DoF + (size_t)(b * T + t) * 16;
      float m[16];
      for (int i = 0; i < 12; ++i) m[i] = d[i];
      m[12] = 0.f; m[13] = 0.f; m[14] = 0.f; m[15] = 1.f;
      float nc[16];
      for (int i = 0; i < 4; ++i)
        for (int j = 0; j < 4; ++j) {
          float s = 0.f;
          for (int k = 0; k < 4; ++k) s = fmaf(m[i * 4 + k], c[k * 4 + j], s);
          nc[i * 4 + j] = s;
        }
      for (int i = 0; i < 16; ++i) c[i] = nc[i];
    }
    for (int i = 0; i < 12; ++i) cum[(size_t)(b * T + t) * 12 + i] = c[i];
  }
}

// ---------------------------------------------------------------------------
// Kernel 1: frustum projection + trilinear sample -> LDS A[16x96],
// then W_N GEMM via V_WMMA_F32_16X16X4_F32, bias + ReLU -> h workspace.
// h layout: [b*T+t][point][40] (channel-contiguous for stage-3 gathers).
// ---------------------------------------------------------------------------
__global__ __launch_bounds__(128) void k_sample_gemmN(
    const float* __restrict__ feat, const float* __restrict__ RT,
    const float* __restrict__ intr, const float* __restrict__ WN,
    const float* __restrict__ bN, float* __restrict__ hbuf) {
  __shared__ float Alds[16][K1 + 4];
  const int blk = blockIdx.x;
  const int ptile = blk % PTILES;
  const int bt = blk / PTILES;      // b*T + t
  const int b = bt / T;
  const int tid = threadIdx.x;

  // Phase 1: 96 threads, one (voxel-in-tile, camera) pair each.
  if (tid < 16 * NCAM) {
    const int m = tid & 15;
    const int cam = tid >> 4;
    const int pidx = ptile * 16 + m;
    const int x = pidx / (GZ * GY);
    const int rr = pidx % (GZ * GY);
    const int zz = rr / GY;
    const int yy = rr % GY;
    // voxel centers (vs = 1.0 in all three axes for these bounds)
    const float px = -50.0f + (float)x + 0.5f;
    const float py = -50.0f + (float)yy + 0.5f;
    const float pz = -3.0f + (float)zz + 0.5f;
    const float* rt = RT + (size_t)(bt * NCAM + cam) * 16;
    const float cx = rt[0]*px + rt[1]*py + rt[2]*pz + rt[3];
    const float cy = rt[4]*px + rt[5]*py + rt[6]*pz + rt[7];
    const float cz = rt[8]*px + rt[9]*py + rt[10]*pz + rt[11];
    const float* Km = intr + (size_t)b * 9;
    const float qx = Km[0]*cx + Km[1]*cy + Km[2]*cz;
    const float qy = Km[3]*cx + Km[4]*cy + Km[5]*cz;
    const float qz = Km[6]*cx + Km[7]*cy + Km[8]*cz;
    const float zc = fmaxf(qz, 1e-5f);
    // align_corners=True makes (g+1)*0.5*(dim-1) collapse back to u,v,dbin
    const float u = qx / zc;
    const float v = qy / zc;
    const float dbin = (qz - 2.0f) * (48.0f / 44.8f);

    const float fx0 = floorf(u), fy0 = floorf(v), fz0 = floorf(dbin);
    const float wx1 = u - fx0, wy1 = v - fy0, wz1 = dbin - fz0;
    const float wx0 = 1.f - wx1, wy0 = 1.f - wy1, wz0 = 1.f - wz1;
    int offs[8];
    float wts[8];
#pragma unroll
    for (int j = 0; j < 8; ++j) {
      const int jx = j & 1, jy = (j >> 1) & 1, jz = (j >> 2) & 1;
      const float cxf = fx0 + (float)jx;
      const float cyf = fy0 + (float)jy;
      const float czf = fz0 + (float)jz;
      const bool valid = (cxf >= 0.f) && (cxf <= (float)(FW - 1)) &&
                         (cyf >= 0.f) && (cyf <= (float)(FH - 1)) &&
                         (czf >= 0.f) && (czf <= (float)(FD - 1));
      const int xi = (int)fminf(fmaxf(cxf, 0.f), (float)(FW - 1));
      const int yi = (int)fminf(fmaxf(cyf, 0.f), (float)(FH - 1));
      const int zi = (int)fminf(fmaxf(czf, 0.f), (float)(FD - 1));
      offs[j] = (zi * FH + yi) * FW + xi;
      const float w =
          (jx ? wx1 : wx0) * (jy ? wy1 : wy0) * (jz ? wz1 : wz0);
      wts[j] = valid ? w : 0.f;
    }
    const float* fb = feat + (size_t)(bt * NCAM + cam) * C * FSTRIDE;
#pragma unroll
    for (int c = 0; c < C; ++c) {
      const float* fc = fb + (size_t)c * FSTRIDE;
      float acc = 0.f;
#pragma unroll
      for (int j = 0; j < 8; ++j) acc = fmaf(wts[j], fc[offs[j]], acc);
      Alds[m][cam * C + c] = acc;   // channel index = n*C + c (N*C reshape)
    }
  }
  __syncthreads();

  // Phase 2: waves 0..2 each own a 16-wide N tile of the 40 output channels.
  const int wv = tid >> 5;
  const int lane = tid & 31;
  if (wv < 3) {
    const int m = lane & 15;              // A row / B column
    const int khalf = (lane >> 4) << 1;   // low lanes: K 0,1  high lanes: K 2,3
    const int o = wv * 16 + m;
    const float msk = (o < OUT_CH) ? 1.f : 0.f;
    const float* wrow = WN + (size_t)(o < OUT_CH ? o : 0) * K1;
    v8f acc = {};
#pragma unroll
    for (int k = 0; k < K1; k += 4) {
      v2f a, bm;
      a.x = Alds[m][k + khalf];
      a.y = Alds[m][k + khalf + 1];
      bm.x = wrow[k + khalf] * msk;       // B[k][n] = W_N[n][k]
      bm.y = wrow[k + khalf + 1] * msk;
      acc = __builtin_amdgcn_wmma_f32_16x16x4_f32(
          false, a, false, bm, (short)0, acc, false, false);
    }
    if (o < OUT_CH) {
      const float bias = bN[o];
      const int mbase = (lane >> 4) << 3;  // high lanes carry M = i+8
#pragma unroll
      for (int i = 0; i < 8; ++i) {
        const int pidx = ptile * 16 + mbase + i;
        hbuf[((size_t)bt * NPT + pidx) * OUT_CH + o] =
            fmaxf(acc[i] + bias, 0.f);
      }
    }
  }
}

// ---------------------------------------------------------------------------
// Kernel 2: affine warp + trilinear sample of h -> LDS A[16x120],
// then W_T GEMM via WMMA, bias + ReLU -> output [B,40,GX,GZ,GY].
// ---------------------------------------------------------------------------
__global__ __launch_bounds__(128) void k_warp_gemmT(
    const float* __restrict__ hbuf, const float* __restrict__ cum,
    const float* __restrict__ WT, const float* __restrict__ bT,
    float* __restrict__ out) {
  __shared__ float Alds[16][K2 + 4];
  const int blk = blockIdx.x;
  const int ptile = blk % PTILES;
  const int b = blk / PTILES;
  const int tid = threadIdx.x;

  // Phase 1: 48 threads, one (point-in-tile, t) pair each.
  if (tid < 16 * T) {
    const int m = tid & 15;
    const int t = tid >> 4;
    const int pidx = ptile * 16 + m;
    const int d = pidx / (GZ * GY);     // D axis = GX
    const int rr = pidx % (GZ * GY);
    const int hh = rr / GY;             // H axis = GZ
    const int w = rr % GY;              // W axis = GY
    const float xb = -1.f + (float)w * (2.f / (float)(GY - 1));
    const float yb = -1.f + (float)hh * (2.f / (float)(GZ - 1));
    const float zb = -1.f + (float)d * (2.f / (float)(GX - 1));
    const float* cm = cum + (size_t)(b * T + t) * 12;
    const float gx = cm[0]*xb + cm[1]*yb + cm[2]*zb + cm[3];
    const float gy = cm[4]*xb + cm[5]*yb + cm[6]*zb + cm[7];
    const float gz = cm[8]*xb + cm[9]*yb + cm[10]*zb + cm[11];
    const float ixf = (gx + 1.f) * 0.5f * (float)(GY - 1);
    const float iyf = (gy + 1.f) * 0.5f * (float)(GZ - 1);
    const float izf = (gz + 1.f) * 0.5f * (float)(GX - 1);
    const float fx0 = floorf(ixf), fy0 = floorf(iyf), fz0 = floorf(izf);
    const float wx1 = ixf - fx0, wy1 = iyf - fy0, wz1 = izf - fz0;
    const float wx0 = 1.f - wx1, wy0 = 1.f - wy1, wz0 = 1.f - wz1;
    int offs[8];
    float wts[8];
#pragma unroll
    for (int j = 0; j < 8; ++j) {
      const int jx = j & 1, jy = (j >> 1) & 1, jz = (j >> 2) & 1;
      const float cxf = fx0 + (float)jx;
      const float cyf = fy0 + (float)jy;
      const float czf = fz0 + (float)jz;
      const bool valid = (cxf >= 0.f) && (cxf <= (float)(GY - 1)) &&
                         (cyf >= 0.f) && (cyf <= (float)(GZ - 1)) &&
                         (czf >= 0.f) && (czf <= (float)(GX - 1));
      const int xi = (int)fminf(fmaxf(cxf, 0.f), (float)(GY - 1));
      const int yi = (int)fminf(fmaxf(cyf, 0.f), (float)(GZ - 1));
      const int zi = (int)fminf(fmaxf(czf, 0.f), (float)(GX - 1));
      offs[j] = ((zi * GZ + yi) * GY + xi) * OUT_CH;
      const float ww =
          (jx ? wx1 : wx0) * (jy ? wy1 : wy0) * (jz ? wz1 : wz0);
      wts[j] = valid ? ww : 0.f;
    }
    const float* hb = hbuf + (size_t)(b * T + t) * NPT * OUT_CH;
    for (int ch = 0; ch < OUT_CH; ++ch) {
      float acc = 0.f;
#pragma unroll
      for (int j = 0; j < 8; ++j) acc = fmaf(wts[j], hb[offs[j] + ch], acc);
      Alds[m][t * OUT_CH + ch] = acc;   // channel = t*40 + ch
    }
  }
  __syncthreads();

  const int wv = tid >> 5;
  const int lane = tid & 31;
  if (wv < 3) {
    const int m = lane & 15;
    const int khalf = (lane >> 4) << 1;
    const int o = wv * 16 + m;
    const float msk = (o < OUT_CH) ? 1.f : 0.f;
    const float* wrow = WT + (size_t)(o < OUT_CH ? o : 0) * K2;
    v8f acc = {};
#pragma unroll
    for (int k = 0; k < K2; k += 4) {
      v2f a, bm;
      a.x = Alds[m][k + khalf];
      a.y = Alds[m][k + khalf + 1];
      bm.x = wrow[k + khalf] * msk;
      bm.y = wrow[k + khalf + 1] * msk;
      acc = __builtin_amdgcn_wmma_f32_16x16x4_f32(
          false, a, false, bm, (short)0, acc, false, false);
    }
    if (o < OUT_CH) {
      const float bias = bT[o];
      const int mbase = (lane >> 4) << 3;
#pragma unroll
      for (int i = 0; i < 8; ++i) {
        const int pidx = ptile * 16 + mbase + i;
        out[((size_t)(b * OUT_CH + o)) * NPT + pidx] =
            fmaxf(acc[i] + bias, 0.f);
      }
    }
  }
}

// ---------------------------------------------------------------------------
extern "C" void kernel_launch(void* const* d_in, const int* in_sizes, int n_in,
                              void* d_out, int out_size, void* d_ws,
                              size_t ws_size, hipStream_t stream) {
  (void)in_sizes; (void)n_in; (void)out_size; (void)ws_size;
  const float* feat = (const float*)d_in[0];  // frustum_features
  const float* RT   = (const float*)d_in[1];
  const float* intr = (const float*)d_in[2];
  const float* DoF  = (const float*)d_in[3];
  const float* WN   = (const float*)d_in[4];
  const float* bN   = (const float*)d_in[5];
  const float* WT   = (const float*)d_in[6];
  const float* bT   = (const float*)d_in[7];
  float* out = (float*)d_out;

  float* ws = (float*)d_ws;
  float* cum = ws;              // B*T*12 = 72 floats
  float* hbuf = ws + 256;       // B*T*NPT*OUT_CH = 19.2M floats (76.8 MB)

  k_cum<<<1, 32, 0, stream>>>(DoF, cum);
  k_sample_gemmN<<<B * T * PTILES, 128, 0, stream>>>(feat, RT, intr, WN, bN,
                                                     hbuf);
  k_warp_gemmT<<<B * PTILES, 128, 0, stream>>>(hbuf, cum, WT, bT, out);
}